// GraphPreProcessingLayer_2216203125145
// MI455X (gfx1250) — compile-verified
//
#include <hip/hip_runtime.h>

typedef __attribute__((ext_vector_type(2))) float v2f;
typedef __attribute__((ext_vector_type(8))) float v8f;

// ---------------------------------------------------------------------------
// Kernel 1: rowsum[band*16 + r] += sum_k adj[row, kchunk] via WMMA f32 16x16x4
// One wave (32 lanes) per block. Each wave reduces a 16-row band over a
// K-chunk of the matrix. A-matrix layout (16x4 f32): lane L in [0,16) holds
// row L, K = k0,k0+1 ; lane L+16 holds row L, K = k0+2,k0+3  -> one b64 load
// per lane per WMMA. B = all-ones => D[m,n] = sum_k A[m,k] (every column).
// ---------------------------------------------------------------------------
__global__ __launch_bounds__(32) void rowsum_wmma_kernel(
    const float* __restrict__ adj, float* __restrict__ rowsum,
    int N, int kchunk_len) {
  const int band = blockIdx.x;            // 16-row band index
  const int lane = threadIdx.x;           // 0..31
  const int row  = lane & 15;
  const int koff = (lane >> 4) << 1;      // 0 for lanes 0-15, 2 for lanes 16-31
  const size_t kbase = (size_t)blockIdx.y * (size_t)kchunk_len;

  const float* base = adj + (size_t)(band * 16 + row) * (size_t)N + kbase + koff;

  v8f acc = {};                           // 16x16 f32 accumulator (8 VGPRs)
  const v2f ones = {1.0f, 1.0f};

  // kchunk_len is a multiple of 16 -> unroll 4 WMMAs per iteration so the
  // memory pipeline has several b64 loads in flight per wave.
  for (int k = 0; k < kchunk_len; k += 16) {
    v2f a0 = *(const v2f*)(base + k + 0);
    v2f a1 = *(const v2f*)(base + k + 4);
    v2f a2 = *(const v2f*)(base + k + 8);
    v2f a3 = *(const v2f*)(base + k + 12);
    acc = __builtin_amdgcn_wmma_f32_16x16x4_f32(false, a0, false, ones,
                                                (short)0, acc, false, false);
    acc = __builtin_amdgcn_wmma_f32_16x16x4_f32(false, a1, false, ones,
                                                (short)0, acc, false, false);
    acc = __builtin_amdgcn_wmma_f32_16x16x4_f32(false, a2, false, ones,
                                                (short)0, acc, false, false);
    acc = __builtin_amdgcn_wmma_f32_16x16x4_f32(false, a3, false, ones,
                                                (short)0, acc, false, false);
  }

  // C/D layout: lane n (n<16) holds column n of rows 0..7 in acc[0..7];
  // lane n+16 holds column n of rows 8..15. All columns are identical
  // (B was all ones), so lanes 0 and 16 publish the 16 row sums.
  if ((lane & 15) == 0) {
    const int rbase = band * 16 + (lane >> 4) * 8;
#pragma unroll
    for (int r = 0; r < 8; ++r) {
      atomicAdd(&rowsum[rbase + r], acc[r]);
    }
  }
}

// ---------------------------------------------------------------------------
// Kernel 2: dis[i] = rsqrt(rowsum[i] - adj[i,i] + 1)   (diagonal set to 1)
// ---------------------------------------------------------------------------
__global__ void finalize_dis_kernel(const float* __restrict__ adj,
                                    const float* __restrict__ rowsum,
                                    float* __restrict__ dis, int N) {
  const int i = blockIdx.x * blockDim.x + threadIdx.x;
  if (i < N) {
    float s = rowsum[i] - adj[(size_t)i * (size_t)N + i] + 1.0f;
    dis[i] = rsqrtf(s);
  }
}

// ---------------------------------------------------------------------------
// Kernel 3: out[i,j] = (i==j ? 1 : adj[i,j]) * dis[i] * dis[j]
// float4-vectorized, fully coalesced; dis[i] is block-uniform (scalar load).
// ---------------------------------------------------------------------------
__global__ __launch_bounds__(256) void scale_kernel(
    const float* __restrict__ adj, const float* __restrict__ dis,
    float* __restrict__ out, int N) {
  const int i  = blockIdx.y;
  const int j4 = blockIdx.x * blockDim.x + threadIdx.x;
  const int j  = j4 << 2;
  const size_t rowoff = (size_t)i * (size_t)N;

  float4 a  = reinterpret_cast<const float4*>(adj + rowoff)[j4];
  const float di = dis[i];
  float4 dj = reinterpret_cast<const float4*>(dis)[j4];

  if ((unsigned)(i - j) < 4u) {           // diagonal element lives in this quad
    reinterpret_cast<float*>(&a)[i - j] = 1.0f;
  }

  float4 o;
  o.x = a.x * di * dj.x;
  o.y = a.y * di * dj.y;
  o.z = a.z * di * dj.z;
  o.w = a.w * di * dj.w;
  reinterpret_cast<float4*>(out + rowoff)[j4] = o;
}

// ---------------------------------------------------------------------------
extern "C" void kernel_launch(void* const* d_in, const int* in_sizes, int n_in,
                              void* d_out, int out_size, void* d_ws, size_t ws_size,
                              hipStream_t stream) {
  const float* adj = (const float*)d_in[0];
  float* out = (float*)d_out;

  // N = sqrt(in_sizes[0]) (N=8192 for the reference inputs)
  int N = 1;
  while ((long long)N * (long long)N < (long long)in_sizes[0]) N <<= 1;

  float* rowsum = (float*)d_ws;        // N floats
  float* dis    = rowsum + N;          // N floats

  // Zero rowsum accumulator (workspace is poisoned, atomics need zeros).
  hipMemsetAsync(rowsum, 0, (size_t)N * sizeof(float), stream);

  // Pass 1: WMMA rowsum. 16-row bands x 8 K-splits => N/16 * 8 single-wave blocks.
  const int K_SPLIT = 8;
  const int kchunk_len = N / K_SPLIT;  // 1024 for N=8192 (multiple of 16)
  dim3 g1(N / 16, K_SPLIT);
  rowsum_wmma_kernel<<<g1, 32, 0, stream>>>(adj, rowsum, N, kchunk_len);

  // Pass 2: d^{-1/2} with diagonal correction.
  finalize_dis_kernel<<<(N + 255) / 256, 256, 0, stream>>>(adj, rowsum, dis, N);

  // Pass 3: symmetric scaling, float4 per thread.
  dim3 g3((N / 4) / 256, N);
  scale_kernel<<<g3, 256, 0, stream>>>(adj, dis, out, N);
}